// Model_12773232738731
// MI455X (gfx1250) — compile-verified
//
#include <hip/hip_runtime.h>
#include <hip/hip_bf16.h>

typedef __attribute__((ext_vector_type(2))) float v2f;
typedef __attribute__((ext_vector_type(8))) float v8f;

// ---------------------------------------------------------------------------
// Degree / normalization
// ---------------------------------------------------------------------------
__global__ void k_init_deg(float* __restrict__ deg, int n) {
    int i = blockIdx.x * blockDim.x + threadIdx.x;
    if (i < n) deg[i] = 1.0f;  // self-loop contributes 1 to every node's degree
}

__global__ void k_deg_scatter(const int* __restrict__ dst, float* __restrict__ deg, int E) {
    int e = blockIdx.x * blockDim.x + threadIdx.x;
    if (e < E) unsafeAtomicAdd(&deg[dst[e]], 1.0f);  // global_atomic_add_f32, non-returning
}

__global__ void k_deg_to_dinv(float* __restrict__ deg, int n) {
    int i = blockIdx.x * blockDim.x + threadIdx.x;
    if (i < n) deg[i] = __frsqrt_rn(deg[i]);  // deg >= 1 always (self-loop)
}

// ---------------------------------------------------------------------------
// WMMA GEMM, fp32 path: V_WMMA_F32_16X16X4_F32, one wave per 16-row tile.
// A (16x4 f32): lanes 0-15 hold M=0..15 {K=kb,kb+1}; lanes 16-31 {K=kb+2,kb+3}
// B (4x16 f32): lanes 0-15 hold N=0..15 rows {kb,kb+1}; lanes 16-31 {kb+2,kb+3}
// D (16x16 f32): 8 VGPRs; lanes 0-15 -> M=v, lanes 16-31 -> M=v+8, N=lane&15
// ---------------------------------------------------------------------------
__global__ __launch_bounds__(256) void k_gemm1_wmma(
    const float* __restrict__ X,   // [N,128] row-major
    const float* __restrict__ W,   // [128,16] row-major
    float* __restrict__ out,       // [N,16]
    int n)
{
    const int lane  = threadIdx.x & 31;
    const int wave  = threadIdx.x >> 5;
    const int tile  = blockIdx.x * 8 + wave;
    const int row0  = tile * 16;
    if (row0 + 16 > n) return;                 // wave-uniform exit; EXEC all-1 for WMMA
    const int m     = lane & 15;
    const int khalf = (lane >> 4) << 1;        // 0 or 2
    const float* arow = X + (size_t)(row0 + m) * 128;

    v8f acc = {};
    #pragma unroll
    for (int kb = 0; kb < 128; kb += 4) {
        v2f a, b;
        a.x = arow[kb + khalf];
        a.y = arow[kb + khalf + 1];
        b.x = W[(kb + khalf)     * 16 + m];
        b.y = W[(kb + khalf + 1) * 16 + m];
        acc = __builtin_amdgcn_wmma_f32_16x16x4_f32(
            false, a, false, b, (short)0, acc, false, false);
    }
    const int rsel = (lane >> 4) << 3;         // 0 or 8
    #pragma unroll
    for (int v = 0; v < 8; ++v)
        out[(size_t)(row0 + v + rsel) * 16 + m] = acc[v];
}

__global__ __launch_bounds__(256) void k_gemm2_wmma(
    const float* __restrict__ H,   // [N,16] row-major
    const float* __restrict__ W,   // [16,8] row-major (zero-padded to 16 cols)
    float* __restrict__ out,       // [N,8]
    int n)
{
    const int lane  = threadIdx.x & 31;
    const int wave  = threadIdx.x >> 5;
    const int tile  = blockIdx.x * 8 + wave;
    const int row0  = tile * 16;
    if (row0 + 16 > n) return;
    const int m     = lane & 15;
    const int khalf = (lane >> 4) << 1;
    const float* arow = H + (size_t)(row0 + m) * 16;

    v8f acc = {};
    #pragma unroll
    for (int kb = 0; kb < 16; kb += 4) {
        v2f a, b;
        a.x = arow[kb + khalf];
        a.y = arow[kb + khalf + 1];
        b.x = (m < 8) ? W[(kb + khalf)     * 8 + m] : 0.0f;
        b.y = (m < 8) ? W[(kb + khalf + 1) * 8 + m] : 0.0f;
        acc = __builtin_amdgcn_wmma_f32_16x16x4_f32(
            false, a, false, b, (short)0, acc, false, false);
    }
    const int rsel = (lane >> 4) << 3;
    if (m < 8) {
        #pragma unroll
        for (int v = 0; v < 8; ++v)
            out[(size_t)(row0 + v + rsel) * 8 + m] = acc[v];
    }
}

// ---------------------------------------------------------------------------
// h[i][f] = bias[f] + dinv[i]^2 * hw[i][f]   (self-loop message fused in)
// ---------------------------------------------------------------------------
__global__ void k_init_h(const float* __restrict__ hw, const float* __restrict__ dinv,
                         const float* __restrict__ bias, float* __restrict__ h,
                         int n, int logF)
{
    int i = blockIdx.x * blockDim.x + threadIdx.x;
    int total = n << logF;
    if (i >= total) return;
    int node = i >> logF;
    int f    = i & ((1 << logF) - 1);
    float d  = dinv[node];
    h[i] = bias[f] + d * d * hw[i];
}

// ---------------------------------------------------------------------------
// Edge scatter: one thread per (edge, float4 chunk).
// h[dst] += dinv[src]*dinv[dst] * hw[src]; node tables are L2-resident.
// ---------------------------------------------------------------------------
__global__ void k_edge_scatter(const int* __restrict__ src, const int* __restrict__ dst,
                               const float* __restrict__ dinv,
                               const float* __restrict__ hw, float* __restrict__ h,
                               int E, int logF)
{
    int gid = blockIdx.x * blockDim.x + threadIdx.x;
    const int cshift = logF - 2;                 // chunks per edge = F/4
    int total = E << cshift;
    if (gid >= total) return;
    int e  = gid >> cshift;
    int ch = gid & ((1 << cshift) - 1);
    int s  = src[e];
    int d  = dst[e];
    float nrm = dinv[s] * dinv[d];
    const float4 v = *((const float4*)(hw + ((size_t)s << logF)) + ch);
    float* dp = h + ((size_t)d << logF) + (ch << 2);
    unsafeAtomicAdd(dp + 0, nrm * v.x);
    unsafeAtomicAdd(dp + 1, nrm * v.y);
    unsafeAtomicAdd(dp + 2, nrm * v.z);
    unsafeAtomicAdd(dp + 3, nrm * v.w);
}

// ---------------------------------------------------------------------------
// z[i] = sigmoid(h2[i] . Wl + bl)
// ---------------------------------------------------------------------------
__global__ void k_link_z(const float* __restrict__ h2, const float* __restrict__ Wl,
                         const float* __restrict__ bl, float* __restrict__ z, int n)
{
    int i = blockIdx.x * blockDim.x + threadIdx.x;
    if (i >= n) return;
    const float* r = h2 + (size_t)i * 8;
    float s = bl[0];
    #pragma unroll
    for (int j = 0; j < 8; ++j) s += r[j] * Wl[j];
    z[i] = 1.0f / (1.0f + __expf(-s));
}

// Pad W2 [16,8] -> not needed (kernel handles it), but pred gather:
__global__ void k_pred(const int* __restrict__ pe, const float* __restrict__ z,
                       float* __restrict__ out, int P)
{
    int p = blockIdx.x * blockDim.x + threadIdx.x;
    if (p >= P) return;
    out[p] = z[pe[2 * p]] * z[pe[2 * p + 1]];
}

// ---------------------------------------------------------------------------
// Launch
// ---------------------------------------------------------------------------
extern "C" void kernel_launch(void* const* d_in, const int* in_sizes, int n_in,
                              void* d_out, int out_size, void* d_ws, size_t ws_size,
                              hipStream_t stream) {
    const float* x   = (const float*)d_in[0];   // [N,128]
    const int*   ei  = (const int*)d_in[1];     // [2,E] flat: src = ei[0:E), dst = ei[E:2E)
    const int*   pe  = (const int*)d_in[2];     // [P,2]
    const float* W1  = (const float*)d_in[3];   // [128,16]
    const float* b1  = (const float*)d_in[4];   // [16]
    const float* W2  = (const float*)d_in[5];   // [16,8]
    const float* b2  = (const float*)d_in[6];   // [8]
    const float* Wl  = (const float*)d_in[7];   // [8]
    const float* bl  = (const float*)d_in[8];   // [1]

    const int N = in_sizes[0] / 128;
    const int E = in_sizes[1] / 2;
    const int P = in_sizes[2] / 2;
    const int* src = ei;
    const int* dst = ei + E;

    // workspace layout (floats, 64-elt aligned)
    float* ws = (float*)d_ws;
    auto align64 = [](size_t v) { return (v + 63) & ~(size_t)63; };
    size_t off = 0;
    float* dinv = ws + off; off = align64(off + (size_t)N);
    float* hw1  = ws + off; off = align64(off + (size_t)N * 16);
    float* h1   = ws + off; off = align64(off + (size_t)N * 16);
    float* hw2  = ws + off; off = align64(off + (size_t)N * 8);
    float* h2   = ws + off; off = align64(off + (size_t)N * 8);
    float* z    = ws + off; off = align64(off + (size_t)N);
    (void)ws_size;

    const int B = 256;
    const int tiles  = (N + 15) / 16;
    const int gblk   = (tiles + 7) / 8;   // 8 waves per block, 1 tile per wave

    // 1) normalization
    k_init_deg   <<<(N + B - 1) / B, B, 0, stream>>>(dinv, N);
    k_deg_scatter<<<(E + B - 1) / B, B, 0, stream>>>(dst, dinv, E);
    k_deg_to_dinv<<<(N + B - 1) / B, B, 0, stream>>>(dinv, N);

    // 2) layer 1: hw1 = x @ W1 (WMMA f32), then h1 = b1 + selfloop + scatter
    k_gemm1_wmma<<<gblk, B, 0, stream>>>(x, W1, hw1, N);
    k_init_h<<<((N * 16) + B - 1) / B, B, 0, stream>>>(hw1, dinv, b1, h1, N, 4);
    {
        int total = E * 4;  // F=16 -> 4 float4 chunks per edge
        k_edge_scatter<<<(total + B - 1) / B, B, 0, stream>>>(src, dst, dinv, hw1, h1, E, 4);
    }

    // 3) layer 2: hw2 = h1 @ W2 (WMMA f32), then h2 = b2 + selfloop + scatter
    k_gemm2_wmma<<<gblk, B, 0, stream>>>(h1, W2, hw2, N);
    k_init_h<<<((N * 8) + B - 1) / B, B, 0, stream>>>(hw2, dinv, b2, h2, N, 3);
    {
        int total = E * 2;  // F=8 -> 2 float4 chunks per edge
        k_edge_scatter<<<(total + B - 1) / B, B, 0, stream>>>(src, dst, dinv, hw2, h2, E, 3);
    }

    // 4) link predictor
    k_link_z<<<(N + B - 1) / B, B, 0, stream>>>(h2, Wl, bl, z, N);
    k_pred  <<<(P + B - 1) / B, B, 0, stream>>>(pe, z, (float*)d_out, P);
}